// ListFoldLoss_16793322127949
// MI455X (gfx1250) — compile-verified
//
#include <hip/hip_runtime.h>
#include <math.h>

// CDNA5 wave32 / WMMA types
typedef __attribute__((ext_vector_type(2))) float v2f;
typedef __attribute__((ext_vector_type(8))) float v8f;

#define NELEM 8192
#define NT    1024   // threads per block = 32 waves
#define NSEG  512    // 8192 / 16 segments
#define NPAIR 4096

// ---------------------------------------------------------------------------
// Per-wave 256-element segmented inclusive scan using V_WMMA_F32_16X16X4_F32.
// The wave owns s[base .. base+255], viewed as X[r][c] = s[base + r*16 + c].
// Row-wise cumsum  Y = X * U  (U = 16x16 upper-triangular ones), computed as
// four accumulating K=4 WMMAs:  Y = sum_c  X[:,4c:4c+4] * U[4c:4c+4,:].
// A layout (16x4 f32): lane holds row M=lane&15; VGPR0 = K0 (lanes 0-15) /
// K2 (lanes 16-31); VGPR1 = K1/K3.  B (4x16): VGPR0 = row K0/K2 striped over
// lanes (N=lane&15), VGPR1 = row K1/K3.  C/D (16x16 f32): VGPR v holds
// (M = v + 8*(lane>=16), N = lane&15).
// Writes Y back in place and deposits the 16 segment totals into t[].
// Must be called with all 1024 threads active (EXEC all ones for WMMA).
// ---------------------------------------------------------------------------
__device__ __forceinline__ void wave_tile_scan_wmma(float* s, float* t,
                                                    int wave, int lane)
{
    const int base = wave * 256;
    const int m    = lane & 15;        // row for A, column N for B/C/D
    const int hi   = lane >> 4;        // half-wave select

    v8f acc = {};
    #pragma unroll
    for (int c = 0; c < 4; ++c) {
        const int k0 = 4 * c + (hi ? 2 : 0);
        const int k1 = k0 + 1;
        v2f a, b;
        a.x = s[base + m * 16 + k0];             // A = X[:, k-slice]
        a.y = s[base + m * 16 + k1];
        b.x = (k0 <= m) ? 1.0f : 0.0f;           // B = U[k-slice, :]
        b.y = (k1 <= m) ? 1.0f : 0.0f;
        acc = __builtin_amdgcn_wmma_f32_16x16x4_f32(
            /*neg_a=*/false, a, /*neg_b=*/false, b,
            /*c_mod=*/(short)0, acc, /*reuse_a=*/false, /*reuse_b=*/false);
    }

    // Write Y (row-wise inclusive cumsum of the tile) back in place.
    #pragma unroll
    for (int v = 0; v < 8; ++v) {
        const int row = v + (hi ? 8 : 0);
        s[base + row * 16 + m] = acc[v];
    }
    // Lanes holding column N=15 own the segment totals Y[row][15].
    if (m == 15) {
        #pragma unroll
        for (int v = 0; v < 8; ++v) {
            const int row = v + (hi ? 8 : 0);
            t[wave * 16 + row] = acc[v];
        }
    }
}

// Full 8192-element inclusive prefix scan of s[] (t[] is NSEG scratch).
// Caller must __syncthreads() after filling s[] and before calling.
__device__ __forceinline__ void full_scan(float* s, float* t, int tid)
{
    wave_tile_scan_wmma(s, t, tid >> 5, tid & 31);
    __syncthreads();

    // Inclusive Hillis-Steele scan of the 512 segment totals.
    for (int off = 1; off < NSEG; off <<= 1) {
        float v = 0.0f;
        if (tid < NSEG && tid >= off) v = t[tid - off];
        __syncthreads();
        if (tid < NSEG && tid >= off) t[tid] += v;
        __syncthreads();
    }

    // Add exclusive segment offsets -> global inclusive prefix sum.
    #pragma unroll
    for (int e = 0; e < 8; ++e) {
        const int j = tid + e * NT;
        const int g = j >> 4;
        if (g > 0) s[j] += t[g - 1];
    }
    __syncthreads();
}

__global__ __launch_bounds__(NT)
void listfold_loss_kernel(const float* __restrict__ pred,
                          const float* __restrict__ target,
                          float* __restrict__ out)
{
    __shared__ float f[NELEM];   // values (pred), sorted by target desc
    __shared__ float s[NELEM];   // keys during sort, then scan buffer
    __shared__ float t[NSEG];    // segment totals

    const int tid = threadIdx.x;

    // Load: keys = target into s, values = pred into f.
    #pragma unroll
    for (int e = 0; e < 8; ++e) {
        const int i = tid + e * NT;
        s[i] = target[i];
        f[i] = pred[i];
    }

    // Bitonic sort, descending by key, carrying f.
    for (unsigned k = 2; k <= NELEM; k <<= 1) {
        for (unsigned j = k >> 1; j > 0; j >>= 1) {
            __syncthreads();
            #pragma unroll
            for (int e = 0; e < 8; ++e) {
                const unsigned i   = tid + e * NT;
                const unsigned ixj = i ^ j;
                if (ixj > i) {
                    const float ka = s[i], kb = s[ixj];
                    const bool descBlock = ((i & k) == 0);
                    const bool doSwap = descBlock ? (ka < kb) : (ka > kb);
                    if (doSwap) {
                        s[i] = kb; s[ixj] = ka;
                        const float va = f[i]; f[i] = f[ixj]; f[ixj] = va;
                    }
                }
            }
        }
    }
    __syncthreads();

    // ---- Scan #1: inclusive prefix of e^{f} ----
    #pragma unroll
    for (int e = 0; e < 8; ++e) {
        const int i = tid + e * NT;
        s[i] = expf(f[i]);
    }
    __syncthreads();
    full_scan(s, t, tid);

    // P_i = sum_{k=i..8191-i} e^{f_k}; also cache log(num_i) and win_i.
    float Pv[4], lognum[4], winv[4];
    #pragma unroll
    for (int p = 0; p < 4; ++p) {
        const int i = tid + p * NT;                 // i in [0, 4096)
        Pv[p]     = s[8191 - i] - ((i > 0) ? s[i - 1] : 0.0f);
        lognum[p] = f[i] - f[8191 - i];
        winv[p]   = (float)(NELEM - 2 * i);
    }
    __syncthreads();

    // ---- Scan #2: inclusive prefix of e^{-f} ----
    #pragma unroll
    for (int e = 0; e < 8; ++e) {
        const int i = tid + e * NT;
        s[i] = expf(-f[i]);
    }
    __syncthreads();
    full_scan(s, t, tid);

    // Combine: S_i = P_i * Q_i (rank-1 factorization of the D-square sum).
    float partial = 0.0f;
    #pragma unroll
    for (int p = 0; p < 4; ++p) {
        const int i = tid + p * NT;
        const float Q = s[8191 - i] - ((i > 0) ? s[i - 1] : 0.0f);
        float denom = Pv[p] * Q - winv[p];          // S - win * psi(0)
        if (denom <= 0.0f) denom = 1e-8f;
        partial += lognum[p] - logf(denom);
    }
    __syncthreads();

    // Block reduction of 1024 partials in s[0..1023].
    s[tid] = partial;
    __syncthreads();
    for (int str = NT / 2; str > 0; str >>= 1) {
        if (tid < str) s[tid] += s[tid + str];
        __syncthreads();
    }
    if (tid == 0) out[0] = -s[0];
}

extern "C" void kernel_launch(void* const* d_in, const int* in_sizes, int n_in,
                              void* d_out, int out_size, void* d_ws, size_t ws_size,
                              hipStream_t stream)
{
    (void)in_sizes; (void)n_in; (void)out_size; (void)d_ws; (void)ws_size;
    const float* pred   = (const float*)d_in[0];
    const float* target = (const float*)d_in[1];
    float* out = (float*)d_out;
    listfold_loss_kernel<<<dim3(1), dim3(NT), 0, stream>>>(pred, target, out);
}